// GaussianRenderer_46566035423482
// MI455X (gfx1250) — compile-verified
//
#include <hip/hip_runtime.h>
#include <math.h>

typedef float v2f __attribute__((ext_vector_type(2)));
typedef float v8f __attribute__((ext_vector_type(8)));

#define K_ASSIGN 20
#define THR 0.01f
#define TWO_PI 6.28318530717958647692f

// -------- kernel 1: per-gaussian feature records (16 floats each) --------
// layout: [Mm0,Mm1,Mm2,mMm | M00,M01,M02,M10 | M11,M12,M20,M21 | M22,0,0,0]
// mMm sits in the K=3 slot of the dMm B-matrix, which is multiplied by a
// zero A-slot, so it is free to carry as a side-channel.
__global__ void __launch_bounds__(256) gr_prep(
    const float* __restrict__ verts, const float* __restrict__ sigmas,
    const float* __restrict__ origins, float4* __restrict__ G4, int N)
{
  int n = blockIdx.x * blockDim.x + threadIdx.x;
  if (n >= N) return;
  float m0 = verts[n*3+0] - origins[0];
  float m1 = verts[n*3+1] - origins[1];
  float m2 = verts[n*3+2] - origins[2];
  const float* S = sigmas + n*9;
  float M00=2.f*S[0], M01=2.f*S[1], M02=2.f*S[2];
  float M10=2.f*S[3], M11=2.f*S[4], M12=2.f*S[5];
  float M20=2.f*S[6], M21=2.f*S[7], M22=2.f*S[8];
  float Mm0 = M00*m0 + M01*m1 + M02*m2;
  float Mm1 = M10*m0 + M11*m1 + M12*m2;
  float Mm2 = M20*m0 + M21*m1 + M22*m2;
  float mMm = m0*Mm0 + m1*Mm1 + m2*Mm2;
  G4[n*4+0] = make_float4(Mm0, Mm1, Mm2, mMm);
  G4[n*4+1] = make_float4(M00, M01, M02, M10);
  G4[n*4+2] = make_float4(M11, M12, M20, M21);
  G4[n*4+3] = make_float4(M22, 0.f, 0.f, 0.f);
}

// -------- kernel 2: WMMA pair evaluation + top-K + composite --------
// 8 waves / block, each wave owns 16 rays. Per wave LDS region (1536 floats):
//   [0,256)    act tile 16x16 for current chunk
//   [256,576)  top-20 act   (16 rays x 20)
//   [576,896)  top-20 index
//   [896,1216) top-20 depth
//   [1216,1536)top-20 dMd
__global__ void __launch_bounds__(256) gr_render(
    const float* __restrict__ rays, const float4* __restrict__ G4, int N,
    float* __restrict__ out_w, int* __restrict__ out_i,
    int* __restrict__ out_n, float* __restrict__ out_l)
{
  extern __shared__ float smem[];
  const int lane = threadIdx.x & 31;
  const int wave = threadIdx.x >> 5;
  const bool hi  = lane >= 16;
  const int col  = lane & 15;

  float* wbase = smem + wave * 1536;
  float* s_act = wbase;
  float* s_ta  = wbase + 256;
  int*   s_ti  = (int*)(wbase + 576);
  float* s_tl  = wbase + 896;
  float* s_td  = wbase + 1216;

  const int rayBase = (blockIdx.x * 8 + wave) * 16;
  const int myray   = rayBase + col;           // row this lane feeds into A
  const float d0 = rays[myray*3+0];
  const float d1 = rays[myray*3+1];
  const float d2 = rays[myray*3+2];
  const float p0=d0*d0, p1=d0*d1, p2=d0*d2;
  const float p3=d1*d0, p4=d1*d1, p5=d1*d2;
  const float p6=d2*d0, p7=d2*d1, p8=d2*d2;

  // A-matrix lanes: low half carries K=0,1; high half carries K=2,3.
  v2f a_dir, a_q0, a_q1, a_q2;
  if (!hi) { a_dir.x=d0; a_dir.y=d1;  a_q0.x=p0; a_q0.y=p1;
             a_q1.x=p4;  a_q1.y=p5;   a_q2.x=p8; a_q2.y=0.f; }
  else     { a_dir.x=d2; a_dir.y=0.f; a_q0.x=p2; a_q0.y=p3;
             a_q1.x=p6;  a_q1.y=p7;   a_q2.x=0.f;a_q2.y=0.f; }

  int cnt = 0;                       // valid count (lanes 0..15 only)
  const int nchunk = N >> 4;

  for (int ch = 0; ch < nchunk; ++ch) {
    const int g = (ch << 4) + col;
    float4 f0 = G4[g*4+0];
    float4 f1 = G4[g*4+1];
    float4 f2 = G4[g*4+2];
    float4 f3 = G4[g*4+3];
    int gp = g + 16; if (gp >= N) gp -= N;
    __builtin_prefetch((const void*)(G4 + gp*4), 0, 1);

    v2f b_mm, b_q0, b_q1, b_q2;
    if (!hi) { b_mm.x=f0.x; b_mm.y=f0.y; b_q0.x=f1.x; b_q0.y=f1.y;
               b_q1.x=f2.x; b_q1.y=f2.y; b_q2.x=f3.x; b_q2.y=f3.y; }
    else     { b_mm.x=f0.z; b_mm.y=f0.w; b_q0.x=f1.z; b_q0.y=f1.w;
               b_q1.x=f2.z; b_q1.y=f2.w; b_q2.x=f3.z; b_q2.y=f3.w; }
    const float mMm_col = f0.w;

    v8f c_mm = {0,0,0,0,0,0,0,0};
    v8f c_md = {0,0,0,0,0,0,0,0};
    // dMm = d . Mm   (K=3 padded to 4)
    c_mm = __builtin_amdgcn_wmma_f32_16x16x4_f32(
        false, a_dir, false, b_mm, (short)0, c_mm, false, false);
    // dMd = ddT . M9 (K=9 padded to 12 -> three chained K=4 WMMAs)
    c_md = __builtin_amdgcn_wmma_f32_16x16x4_f32(
        false, a_q0, false, b_q0, (short)0, c_md, false, false);
    c_md = __builtin_amdgcn_wmma_f32_16x16x4_f32(
        false, a_q1, false, b_q1, (short)0, c_md, false, false);
    c_md = __builtin_amdgcn_wmma_f32_16x16x4_f32(
        false, a_q2, false, b_q2, (short)0, c_md, false, false);

    #pragma unroll
    for (int r = 0; r < 8; ++r) {
      float dmm = c_mm[r];
      float dmd = c_md[r];
      // dMd > 0 (positive-definite inverse covariance): one v_rcp_f32
      // replaces the IEEE divide sequence; sign(hit_len) == sign(dmm).
      float rd  = __builtin_amdgcn_rcpf(dmd);
      float q   = mMm_col - dmm * dmm * rd;
      float a   = __expf(-0.5f * q);
      a = (dmm > 0.f && a > THR) ? a : 0.f;
      int rr = hi ? (r + 8) : r;
      s_act[rr * 16 + col] = a;
    }
    __syncthreads();
    if (lane < 16) {
      float* ta = s_ta + lane * K_ASSIGN;
      int*   ti = s_ti + lane * K_ASSIGN;
      for (int c = 0; c < 16; ++c) {
        float a = s_act[lane * 16 + c];
        float cut = (cnt == K_ASSIGN) ? ta[K_ASSIGN - 1] : THR;
        if (a > cut) {
          int pos = (cnt < K_ASSIGN) ? cnt : (K_ASSIGN - 1);
          while (pos > 0 && ta[pos - 1] < a) {
            ta[pos] = ta[pos - 1]; ti[pos] = ti[pos - 1]; --pos;
          }
          ta[pos] = a; ti[pos] = (ch << 4) + c;
          if (cnt < K_ASSIGN) ++cnt;
        }
      }
    }
    __syncthreads();
  }

  if (lane < 16) {
    float* ta = s_ta + lane * K_ASSIGN;
    int*   ti = s_ti + lane * K_ASSIGN;
    float* tl = s_tl + lane * K_ASSIGN;
    float* td = s_td + lane * K_ASSIGN;
    // recompute depth + dMd for the <=20 survivors
    for (int j = 0; j < cnt; ++j) {
      int idx = ti[j];
      float4 h0 = G4[idx*4+0];
      float4 h1 = G4[idx*4+1];
      float4 h2 = G4[idx*4+2];
      float4 h3 = G4[idx*4+3];
      float dmm = d0*h0.x + d1*h0.y + d2*h0.z;
      float dmd = p0*h1.x + p1*h1.y + p2*h1.z + p3*h1.w
                + p4*h2.x + p5*h2.y + p6*h2.z + p7*h2.w
                + p8*h3.x;
      tl[j] = dmm * __builtin_amdgcn_rcpf(dmd);
      td[j] = dmd;
    }
    // depth sort ascending (stable insertion, <=20 elements)
    for (int j = 1; j < cnt; ++j) {
      float la = ta[j]; int li = ti[j]; float ll = tl[j]; float ld = td[j];
      int k = j;
      while (k > 0 && tl[k - 1] > ll) {
        ta[k]=ta[k-1]; ti[k]=ti[k-1]; tl[k]=tl[k-1]; td[k]=td[k-1]; --k;
      }
      ta[k]=la; ti[k]=li; tl[k]=ll; td[k]=ld;
    }
    // alpha composite (TAU = 1)
    float cum = 0.f;
    const int base = myray * K_ASSIGN;
    for (int j = 0; j < K_ASSIGN; ++j) {
      if (j < cnt) {
        float alpha = ta[j] * sqrtf(TWO_PI * __builtin_amdgcn_rcpf(td[j]));
        float w = __expf(-cum) * alpha;
        cum += alpha;
        out_w[base + j] = w;
        out_i[base + j] = ti[j];
        out_l[base + j] = tl[j];
      } else {
        out_w[base + j] = 0.f;
        out_i[base + j] = -1;
        out_l[base + j] = 0.f;
      }
    }
    out_n[myray] = cnt;
  }
}

extern "C" void kernel_launch(void* const* d_in, const int* in_sizes, int n_in,
                              void* d_out, int out_size, void* d_ws, size_t ws_size,
                              hipStream_t stream) {
  const float* verts   = (const float*)d_in[0];
  const float* sigmas  = (const float*)d_in[1];
  const float* rays    = (const float*)d_in[2];
  const float* origins = (const float*)d_in[3];
  const int N     = in_sizes[1] / 9;   // gaussians
  const int NRAYS = in_sizes[2] / 3;   // B*H*W

  float4* G4 = (float4*)d_ws;          // N * 16 floats
  gr_prep<<<(N + 255) / 256, 256, 0, stream>>>(verts, sigmas, origins, G4, N);

  float* out_w = (float*)d_out;                                // [NRAYS,20]
  int*   out_i = (int*)(out_w + (size_t)NRAYS * K_ASSIGN);     // [NRAYS,20]
  int*   out_n = (int*)(out_w + (size_t)2 * NRAYS * K_ASSIGN); // [NRAYS]
  float* out_l = out_w + (size_t)2 * NRAYS * K_ASSIGN + NRAYS; // [NRAYS,20]

  const int raysPerBlock = 8 * 16;
  dim3 grid(NRAYS / raysPerBlock);
  size_t smem = 8 * 1536 * sizeof(float);
  gr_render<<<grid, 256, smem, stream>>>(rays, G4, N, out_w, out_i, out_n, out_l);
}